// NodeEncoder_16836271800691
// MI455X (gfx1250) — compile-verified
//
#include <hip/hip_runtime.h>
#include <math.h>

#define N_NODES   8192
#define ATOM_EMB  64
#define MUL_OUT   32
#define NUM_BASIS 32
#define OUT_DIM   288          // 32*1 + 32*3 + 32*5
#define EDGES_PER_WG 128
#define THREADS   256
#define KTOT      2048         // 32 (basis) * 64 (emb)
#define NCOLS     96           // 3 * 32

typedef __attribute__((ext_vector_type(2))) float v2f;
typedef __attribute__((ext_vector_type(8))) float v8f;

// ---------------------------------------------------------------- zero output
__global__ void zero_out_kernel(float* __restrict__ out, int n) {
    int i = blockIdx.x * blockDim.x + threadIdx.x;
    if (i < n) out[i] = 0.0f;
}

// ------------------------------------------------- x = z @ atom_w + atom_b
__global__ void node_embed_kernel(const float* __restrict__ z,
                                  const float* __restrict__ atom_w,
                                  const float* __restrict__ atom_b,
                                  float* __restrict__ x) {
    int i = blockIdx.x * blockDim.x + threadIdx.x;     // n*64 + u
    if (i >= N_NODES * ATOM_EMB) return;
    int n = i >> 6, u = i & 63;
    float acc = atom_b[u];
#pragma unroll
    for (int c = 0; c < 4; ++c)
        acc += z[n * 4 + c] * atom_w[c * ATOM_EMB + u];
    x[i] = acc;
}

// --------- Rt[col*2048 + k*64 + u] = rad_w2[k][ (col/32)*2048 + u*32 + col%32 ]
// Makes B-matrix K-contiguous so WMMA B fragments are single b64 loads.
__global__ void reorg_w2_kernel(const float* __restrict__ rad_w2,
                                float* __restrict__ Rt) {
    int i = blockIdx.x * blockDim.x + threadIdx.x;     // col*2048 + K
    if (i >= NCOLS * KTOT) return;
    int c = i >> 11, K = i & 2047;
    int k = K >> 6, u = K & 63;
    Rt[i] = rad_w2[k * 6144 + (c >> 5) * 2048 + u * 32 + (c & 31)];
}

// ---------------------------------------------------------------- edge kernel
__global__ void __launch_bounds__(THREADS)
edge_kernel(const float* __restrict__ pos,
            const int*   __restrict__ eidx,      // [2, E] flat
            int E,
            const float* __restrict__ rad_w1,    // [32, 32]
            const float* __restrict__ x,         // [8192, 64]
            const float* __restrict__ Rt,        // [96, 2048]
            float* __restrict__ out)             // [8192, 288]
{
    __shared__ float s_rw1[NUM_BASIS * 32];              //  4.0 KB
    __shared__ float s_sph[EDGES_PER_WG * 9];            //  4.5 KB
    __shared__ float s_h  [EDGES_PER_WG * NUM_BASIS];    // 16.0 KB
    __shared__ float s_x  [EDGES_PER_WG * ATOM_EMB];     // 32.0 KB
    __shared__ int   s_dst[EDGES_PER_WG];                //  0.5 KB

    const int tid   = threadIdx.x;
    const int ebase = blockIdx.x * EDGES_PER_WG;

    for (int i = tid; i < NUM_BASIS * 32; i += THREADS) s_rw1[i] = rad_w1[i];
    __syncthreads();

    // ---------------- phase 1: per-edge scalar work (sph, bessel, radial MLP)
    const float C_SQ3  = 1.7320508075688772f;
    const float C_SQ15 = 3.8729833462074170f;   // sqrt(5)*sqrt(3)
    const float C_SQ5  = 2.2360679774997896f;
    const float C_HALF15 = 1.9364916731037085f; // sqrt(5)*sqrt(3)/2
    const float CUT   = 2.5f;
    const float S2C   = 0.8944271909999159f;    // sqrt(2/2.5)
    const float PI_C  = 1.2566370614359172f;    // pi / 2.5
    const float INV_SQRT32 = 0.17677669529663687f;
    const float GAIN  = 1.679f;

    for (int el = tid; el < EDGES_PER_WG; el += THREADS) {
        int e = ebase + el;
        if (e < E) {
            int s = eidx[e];
            int d = eidx[E + e];
            s_dst[el] = d;
            float vx = pos[3 * s + 0] - pos[3 * d + 0];
            float vy = pos[3 * s + 1] - pos[3 * d + 1];
            float vz = pos[3 * s + 2] - pos[3 * d + 2];
            float r2 = vx * vx + vy * vy + vz * vz;
            float r  = sqrtf(r2);
            float inv = (r > 0.0f) ? (1.0f / r) : 0.0f;  // r==0 -> vec==0 -> u=0
            float ux = vx * inv, uy = vy * inv, uz = vz * inv;
            float* sp = &s_sph[el * 9];
            sp[0] = 1.0f;
            sp[1] = C_SQ3 * ux;
            sp[2] = C_SQ3 * uy;
            sp[3] = C_SQ3 * uz;
            sp[4] = C_SQ15 * ux * uz;
            sp[5] = C_SQ15 * ux * uy;
            sp[6] = C_SQ5 * (uy * uy - 0.5f * (ux * ux + uz * uz));
            sp[7] = C_SQ15 * uy * uz;
            sp[8] = C_HALF15 * (uz * uz - ux * ux);

            float rb[NUM_BASIS];
            if (r > 0.0f && r < CUT) {
                float invr = S2C / r;
                float tr = PI_C * r;
#pragma unroll
                for (int k = 0; k < NUM_BASIS; ++k)
                    rb[k] = __sinf(tr * (float)(k + 1)) * invr;
            } else {
#pragma unroll
                for (int k = 0; k < NUM_BASIS; ++k) rb[k] = 0.0f;
            }
            for (int j = 0; j < 32; ++j) {
                float a = 0.0f;
#pragma unroll
                for (int k = 0; k < NUM_BASIS; ++k)
                    a += rb[k] * s_rw1[k * 32 + j];
                a *= INV_SQRT32;
                s_h[el * NUM_BASIS + j] = GAIN * a / (1.0f + __expf(-a));
            }
        } else {
            s_dst[el] = 0;
            for (int j = 0; j < 9;  ++j) s_sph[el * 9 + j] = 0.0f;
            for (int j = 0; j < 32; ++j) s_h[el * NUM_BASIS + j] = 0.0f;
        }
    }

    // gather x[src] tiles (zero out-of-range rows so h=0 * x=0, never 0*NaN)
    for (int i = tid; i < EDGES_PER_WG * ATOM_EMB; i += THREADS) {
        int el = i >> 6;
        int e  = ebase + el;
        float v = 0.0f;
        if (e < E) {
            int s = eidx[e];
            v = x[s * ATOM_EMB + (i & 63)];
        }
        s_x[i] = v;
    }
    __syncthreads();

    // ---------------- phase 2: WMMA — 16 edges x 96 cols per wave, K = 2048
    const int lane = tid & 31;
    const int wid  = tid >> 5;           // 8 waves -> 8 x 16 = 128 edges
    const int eb   = wid * 16;
    const int mn   = lane & 15;          // row m for A, col n for B
    const int kp   = (lane >> 4) << 1;   // K sub-offset: lanes 0-15 -> {0,1}, 16-31 -> {2,3}

    v8f acc[6];
#pragma unroll
    for (int nt = 0; nt < 6; ++nt) acc[nt] = (v8f){0,0,0,0,0,0,0,0};

    const float* hrow = &s_h[(eb + mn) * NUM_BASIS];
    const float* xrow = &s_x[(eb + mn) * ATOM_EMB];

    for (int k = 0; k < NUM_BASIS; ++k) {
        float hv = hrow[k];
        const float* RtK = Rt + k * 64;             // + col*2048 + u
        for (int ub = 0; ub < ATOM_EMB; ub += 4) {
            int u0 = ub + kp;                       // even -> 8B aligned
            v2f a;
            a.x = hv * xrow[u0];
            a.y = hv * xrow[u0 + 1];
#pragma unroll
            for (int nt = 0; nt < 6; ++nt) {
                int col = nt * 16 + mn;
                v2f b = *(const v2f*)(RtK + col * 2048 + u0);
                acc[nt] = __builtin_amdgcn_wmma_f32_16x16x4_f32(
                    false, a, false, b, (short)0, acc[nt], false, false);
            }
        }
    }

    // ---------------- phase 3: scale, multiply by sph per l-block, scatter-add
    const float TSCALE = 0.022097086912079608f;   // 1/(sqrt(32)*sqrt(64))
    const int mrow_base = (lane >> 4) * 8;        // D layout: m = v + 8*(lane/16)

#pragma unroll
    for (int nt = 0; nt < 6; ++nt) {
        const int col = nt * 16 + mn;
        const int l   = nt >> 1;                  // uniform per nt (no divergence)
        const int w   = col & 31;
#pragma unroll
        for (int v = 0; v < 8; ++v) {
            int   el = eb + mrow_base + v;
            float t  = acc[nt][v] * TSCALE;
            float* orow = out + s_dst[el] * OUT_DIM;
            const float* sp = &s_sph[el * 9];
            if (l == 0) {
                atomicAdd(&orow[w], t * sp[0]);
            } else if (l == 1) {
                atomicAdd(&orow[32 + w * 3 + 0], t * sp[1]);
                atomicAdd(&orow[32 + w * 3 + 1], t * sp[2]);
                atomicAdd(&orow[32 + w * 3 + 2], t * sp[3]);
            } else {
                atomicAdd(&orow[128 + w * 5 + 0], t * sp[4]);
                atomicAdd(&orow[128 + w * 5 + 1], t * sp[5]);
                atomicAdd(&orow[128 + w * 5 + 2], t * sp[6]);
                atomicAdd(&orow[128 + w * 5 + 3], t * sp[7]);
                atomicAdd(&orow[128 + w * 5 + 4], t * sp[8]);
            }
        }
    }
}

// ---------------------------------------------------------------------------
extern "C" void kernel_launch(void* const* d_in, const int* in_sizes, int n_in,
                              void* d_out, int out_size, void* d_ws, size_t ws_size,
                              hipStream_t stream) {
    const float* z      = (const float*)d_in[0];
    const float* pos    = (const float*)d_in[1];
    const int*   eidx   = (const int*)  d_in[2];
    const float* atom_w = (const float*)d_in[3];
    const float* atom_b = (const float*)d_in[4];
    const float* rad_w1 = (const float*)d_in[5];
    const float* rad_w2 = (const float*)d_in[6];
    float* out = (float*)d_out;

    const int E = in_sizes[2] / 2;

    float* x  = (float*)d_ws;                  // 8192*64 floats = 2 MB
    float* Rt = x + N_NODES * ATOM_EMB;        // 96*2048 floats = 768 KB

    zero_out_kernel<<<(out_size + 255) / 256, 256, 0, stream>>>(out, out_size);
    node_embed_kernel<<<(N_NODES * ATOM_EMB + 255) / 256, 256, 0, stream>>>(
        z, atom_w, atom_b, x);
    reorg_w2_kernel<<<(NCOLS * KTOT + 255) / 256, 256, 0, stream>>>(rad_w2, Rt);

    int nwg = (E + EDGES_PER_WG - 1) / EDGES_PER_WG;
    edge_kernel<<<nwg, THREADS, 0, stream>>>(pos, eidx, E, rad_w1, x, Rt, out);
}